// DGAT_27410481283418
// MI455X (gfx1250) — compile-verified
//
#include <hip/hip_runtime.h>

typedef float v2f __attribute__((ext_vector_type(2)));
typedef float v8f __attribute__((ext_vector_type(8)));

static constexpr int N_    = 100000;
static constexpr int K_    = 10;
static constexpr int VF_   = 128;
static constexpr int FILT_ = 64;
static constexpr int H_    = 3;

// ---------------------------------------------------------------------------
// Phase 1: Z[b][w][h] = (verts_b * mask_b) @ W[h]   via V_WMMA_F32_16X16X4_F32
// grid.y encodes combo c = b*6 + h*2 + w  (b: 0=int,1=nh; w: 0=Zc,1=Zn)
// One wave -> 32 nodes (two 16-row strips) x 64 filters: B fragments are
// reused across both strips (8 WMMAs per 8 B-loads + 2 A-loads).
// ws layout: 4 arrays of (H,N,FILT) f32.
// ---------------------------------------------------------------------------
__global__ __launch_bounds__(256) void dgat_phase1(
    const float* __restrict__ vint, const float* __restrict__ vnh,
    const int*   __restrict__ is_int,
    const float* __restrict__ Wvc_int, const float* __restrict__ Wvc_nh,
    const float* __restrict__ Wvn_int, const float* __restrict__ Wvn_nh,
    float* __restrict__ ws)
{
    const size_t S = (size_t)H_ * N_ * FILT_;

    int c   = blockIdx.y;        // 0..11
    int b   = c / 6;
    int rem = c % 6;
    int h   = rem >> 1;
    int w   = rem & 1;

    const float* verts = b ? vnh : vint;
    const float* W     = b ? (w ? Wvn_nh : Wvc_nh) : (w ? Wvn_int : Wvc_int);
    W += (size_t)h * VF_ * FILT_;
    const int target = b ? 0 : 1;
    float* zout = ws + (size_t)(b * 2 + w) * S + (size_t)h * N_ * FILT_;

    const int lane = threadIdx.x & 31;
    const int pair = blockIdx.x * 8 + (threadIdx.x >> 5);
    if (pair * 32 >= N_) return;                  // wave-uniform exit
    const int node0 = pair * 32;                  // strip0: node0.., strip1: node0+16..
    const int m     = lane & 15;                  // row (A) / col (B)
    const int sub   = lane >> 4;                  // K half: 0 -> K{0,1}, 1 -> K{2,3}

    const float maskA = (is_int[node0 + m]      == target) ? 1.0f : 0.0f;
    const float maskB = (is_int[node0 + 16 + m] == target) ? 1.0f : 0.0f;
    const float* vrowA = verts + (size_t)(node0 + m)      * VF_;
    const float* vrowB = verts + (size_t)(node0 + 16 + m) * VF_;

    v8f acc0 = {}, acc1 = {}, acc2 = {}, acc3 = {};
    v8f acc4 = {}, acc5 = {}, acc6 = {}, acc7 = {};

    for (int k0 = 0; k0 < VF_; k0 += 4) {
        const int ka = k0 + 2 * sub;
        v2f aA, aB;
        aA.x = maskA * vrowA[ka];
        aA.y = maskA * vrowA[ka + 1];
        aB.x = maskB * vrowB[ka];
        aB.y = maskB * vrowB[ka + 1];
        const float* Wk = W + (size_t)ka * FILT_ + m;
        v2f b0, b1, b2, b3;
        b0.x = Wk[0];   b0.y = Wk[0  + FILT_];
        b1.x = Wk[16];  b1.y = Wk[16 + FILT_];
        b2.x = Wk[32];  b2.y = Wk[32 + FILT_];
        b3.x = Wk[48];  b3.y = Wk[48 + FILT_];
        acc0 = __builtin_amdgcn_wmma_f32_16x16x4_f32(false, aA, false, b0, (short)0, acc0, false, false);
        acc1 = __builtin_amdgcn_wmma_f32_16x16x4_f32(false, aA, false, b1, (short)0, acc1, false, false);
        acc2 = __builtin_amdgcn_wmma_f32_16x16x4_f32(false, aA, false, b2, (short)0, acc2, false, false);
        acc3 = __builtin_amdgcn_wmma_f32_16x16x4_f32(false, aA, false, b3, (short)0, acc3, false, false);
        acc4 = __builtin_amdgcn_wmma_f32_16x16x4_f32(false, aB, false, b0, (short)0, acc4, false, false);
        acc5 = __builtin_amdgcn_wmma_f32_16x16x4_f32(false, aB, false, b1, (short)0, acc5, false, false);
        acc6 = __builtin_amdgcn_wmma_f32_16x16x4_f32(false, aB, false, b2, (short)0, acc6, false, false);
        acc7 = __builtin_amdgcn_wmma_f32_16x16x4_f32(false, aB, false, b3, (short)0, acc7, false, false);
    }

    // D layout: VGPR r, lane l -> row (r + 8*(l>>4)), col (l&15)
#pragma unroll
    for (int r = 0; r < 8; ++r) {
        const int mrow = r + 8 * sub;
        float* zr0 = zout + (size_t)(node0 + mrow)      * FILT_ + m;
        float* zr1 = zout + (size_t)(node0 + 16 + mrow) * FILT_ + m;
        zr0[0]  = acc0[r];
        zr0[16] = acc1[r];
        zr0[32] = acc2[r];
        zr0[48] = acc3[r];
        zr1[0]  = acc4[r];
        zr1[16] = acc5[r];
        zr1[32] = acc6[r];
        zr1[48] = acc7[r];
    }
}

// ---------------------------------------------------------------------------
// Phase 2: gather + masked softmax + aggregate + ReLU.
// One wave per (branch, head, node); lane owns features {2*lane, 2*lane+1}.
// ---------------------------------------------------------------------------
__device__ __forceinline__ float wave_sum(float v) {
#pragma unroll
    for (int off = 16; off >= 1; off >>= 1)
        v += __shfl_xor(v, off, 32);
    return v;
}

__global__ __launch_bounds__(256) void dgat_phase2(
    const int*   __restrict__ nh_idx,   const int*   __restrict__ int_idx,
    const float* __restrict__ nh_edges, const float* __restrict__ int_edges,
    const float* __restrict__ bv_int,   const float* __restrict__ bv_nh,
    const float* __restrict__ a_int,    const float* __restrict__ a_nh,
    const float* __restrict__ ws,       float* __restrict__ out)
{
    const size_t S = (size_t)H_ * N_ * FILT_;

    const int lane = threadIdx.x & 31;
    const int wid  = blockIdx.x * 8 + (threadIdx.x >> 5);
    if (wid >= 2 * H_ * N_) return;               // wave-uniform exit

    const int b   = wid / (H_ * N_);
    const int rem = wid - b * (H_ * N_);
    const int h   = rem / N_;
    const int n   = rem - h * N_;

    const float* Zc   = ws + (size_t)(b * 2 + 0) * S + ((size_t)h * N_ + n) * FILT_;
    const float* ZnB  = ws + (size_t)(b * 2 + 1) * S + (size_t)h * N_ * FILT_;
    const int*   idx  = b ? nh_idx   : int_idx;
    const float* edg  = b ? nh_edges : int_edges;
    const float* av   = (b ? a_nh  : a_int) + (size_t)h * 2 * FILT_;
    const float* bv   = (b ? bv_nh : bv_int) + (size_t)h * FILT_;

    const int f = lane * 2;
    const float2 zc = *reinterpret_cast<const float2*>(Zc + f);
    const float2 a1 = *reinterpret_cast<const float2*>(av + f);
    const float2 a2 = *reinterpret_cast<const float2*>(av + FILT_ + f);

    const float ec = wave_sum(zc.x * a2.x + zc.y * a2.y);

    float2 g[K_];
    float  e[K_];
    float  npart = 0.0f;
#pragma unroll
    for (int k = 0; k < K_; ++k) {
        const int id = idx[(size_t)n * K_ + k];   // wave-uniform
        if (id >= 0) {
            g[k] = *reinterpret_cast<const float2*>(ZnB + (size_t)id * FILT_ + f);
            const float dp = wave_sum(g[k].x * a1.x + g[k].y * a1.y);
            e[k] = (dp + ec) * edg[(size_t)n * K_ + k];
            npart += 1.0f;
        } else {
            g[k].x = 0.0f; g[k].y = 0.0f;
            e[k]   = 0.0f;                        // masked entries keep e=0 (ref semantics)
        }
    }

    // softmax over all K entries (masked ones contribute exp(0 - mx))
    float mx = e[0];
#pragma unroll
    for (int k = 1; k < K_; ++k) mx = fmaxf(mx, e[k]);
    float denom = 0.0f;
#pragma unroll
    for (int k = 0; k < K_; ++k) { e[k] = __expf(e[k] - mx); denom += e[k]; }
    const float inv_d = 1.0f / denom;
    const float inv_n = 1.0f / fmaxf(npart, 1.0f);

    float sx = 0.0f, sy = 0.0f;
#pragma unroll
    for (int k = 0; k < K_; ++k) {
        const float wk = e[k] * inv_d;            // alpha_k; g[k]==0 when masked
        sx += wk * g[k].x;
        sy += wk * g[k].y;
    }
    sx *= inv_n; sy *= inv_n;

    const float2 bb = *reinterpret_cast<const float2*>(bv + f);
    float2 o;
    o.x = fmaxf(zc.x + sx + bb.x, 0.0f);
    o.y = fmaxf(zc.y + sy + bb.y, 0.0f);

    float* op = out + (size_t)b * N_ * H_ * FILT_ + (size_t)n * H_ * FILT_ + (size_t)h * FILT_ + f;
    *reinterpret_cast<float2*>(op) = o;
}

extern "C" void kernel_launch(void* const* d_in, const int* in_sizes, int n_in,
                              void* d_out, int out_size, void* d_ws, size_t ws_size,
                              hipStream_t stream) {
    const float* vertices_int = (const float*)d_in[0];
    const float* vertices_nh  = (const float*)d_in[1];
    const int*   nh_indices   = (const int*)d_in[2];
    const int*   int_indices  = (const int*)d_in[3];
    const float* nh_edges     = (const float*)d_in[4];
    const float* int_edges    = (const float*)d_in[5];
    const int*   is_int       = (const int*)d_in[6];
    const float* Wvc_int      = (const float*)d_in[7];
    const float* Wvc_nh       = (const float*)d_in[8];
    const float* Wvn_int      = (const float*)d_in[9];
    const float* Wvn_nh       = (const float*)d_in[10];
    const float* bv_int       = (const float*)d_in[11];
    const float* bv_nh        = (const float*)d_in[12];
    const float* a_int        = (const float*)d_in[13];
    const float* a_nh         = (const float*)d_in[14];

    float* ws  = (float*)d_ws;
    float* out = (float*)d_out;

    // Phase 1: 12 combos, N/32 = 3125 node-pairs-of-strips, 8 waves per block.
    const int npairs = N_ / 32;
    dim3 g1((npairs + 7) / 8, 12, 1);
    dgat_phase1<<<g1, 256, 0, stream>>>(vertices_int, vertices_nh, is_int,
                                        Wvc_int, Wvc_nh, Wvn_int, Wvn_nh, ws);

    // Phase 2: one wave per (branch, head, node).
    const int totalw = 2 * H_ * N_;
    dgat_phase2<<<(totalw + 7) / 8, 256, 0, stream>>>(nh_indices, int_indices,
                                                      nh_edges, int_edges,
                                                      bv_int, bv_nh, a_int, a_nh,
                                                      ws, out);
}